// MILNCELoss_71150428225985
// MI455X (gfx1250) — compile-verified
//
#include <hip/hip_runtime.h>
#include <math.h>

#define NUM_G4     12000
#define NUM_G8     6000
#define NUM_G      (NUM_G4 + NUM_G8)   // 18000 = 1125 * 16
#define DIMF       128
#define TAU_F      0.5f
#define BETA_F     10.0f

typedef __attribute__((ext_vector_type(2))) float v2f;
typedef __attribute__((ext_vector_type(4))) float v4f;
typedef __attribute__((ext_vector_type(8))) float v8f;

// ---------------------------------------------------------------------------
// Kernel 1: T[g, :] = n[node_idx[g], :] @ W   via V_WMMA_F32_16X16X4_F32
// grid.x = NUM_G/16 = 1125 row tiles, block = 256 (8 waves; wave w = col tile w)
// ---------------------------------------------------------------------------
__global__ void __launch_bounds__(256)
milnce_gemm_t_wmma(const float* __restrict__ nmat,   // [200000,128]
                   const float* __restrict__ Wmat,   // [128,128]
                   const int*   __restrict__ idx4,   // [12000]
                   const int*   __restrict__ idx8,   // [6000]
                   float*       __restrict__ T)      // [18000,128]
{
    const int wave = threadIdx.x >> 5;     // 0..7 -> column tile
    const int lane = threadIdx.x & 31;
    const int l15  = lane & 15;
    const int hi   = lane >> 4;            // 0 for lanes 0-15, 1 for 16-31

    const int row0 = blockIdx.x * 16;
    const int col0 = wave * 16;

    // gathered A row for this lane's M index (lanes 0-15 and 16-31 share M)
    const int g    = row0 + l15;
    const int node = (g < NUM_G4) ? idx4[g] : idx8[g - NUM_G4];
    const float* __restrict__ arow = nmat + (size_t)node * DIMF;

    v8f acc = (v8f)0.0f;

    // K = 128 in steps of 4
    #pragma unroll 4
    for (int kk = 0; kk < DIMF; kk += 4) {
        // A 16x4 f32 layout: VGPR0 = K(2*hi+0), VGPR1 = K(2*hi+1)
        v2f a;
        a.x = arow[kk + 2 * hi + 0];
        a.y = arow[kk + 2 * hi + 1];
        // B 4x16 f32 layout: VGPRv lanes0-15 -> K=v, lanes16-31 -> K=v+2; N = l15
        v2f bfrag;
        bfrag.x = Wmat[(size_t)(kk + 2 * hi + 0) * DIMF + col0 + l15];
        bfrag.y = Wmat[(size_t)(kk + 2 * hi + 1) * DIMF + col0 + l15];

        acc = __builtin_amdgcn_wmma_f32_16x16x4_f32(
            /*neg_a=*/false, a, /*neg_b=*/false, bfrag,
            /*c_mod=*/(short)0, acc, /*reuse_a=*/false, /*reuse_b=*/false);
    }

    // D 16x16 f32 layout: VGPR v -> M = v + 8*hi, N = l15
    #pragma unroll
    for (int v = 0; v < 8; ++v) {
        const int m = v + 8 * hi;
        T[(size_t)(row0 + m) * DIMF + col0 + l15] = acc[v];
    }
}

// ---------------------------------------------------------------------------
// Kernel 2: per-hypernode loss. grid.x = 18000, block = 128 (4 waves).
// Lane l owns float4 elements [4l, 4l+4) of the 128-dim feature vectors, so
// each gathered edge row is one global_load_b128 per lane (512 B / wave).
// ---------------------------------------------------------------------------
__global__ void __launch_bounds__(128)
milnce_loss(const float* __restrict__ T,      // [18000,128]
            const float* __restrict__ emat,   // [200000,128]
            const float* __restrict__ bscal,  // [1]
            const int*   __restrict__ eidx4,  // [12000,4,16]
            const int*   __restrict__ eidx8,  // [6000,8,16]
            float*       __restrict__ lossv)  // [18000]
{
    const int g    = blockIdx.x;
    const int tid  = threadIdx.x;
    const int lane = tid & 31;
    const int wave = tid >> 5;

    int k, L;
    const int* __restrict__ ebase;
    if (g < NUM_G4) { k = 4; L = 64;  ebase = eidx4 + (size_t)g * 64; }
    else            { k = 8; L = 128; ebase = eidx8 + (size_t)(g - NUM_G4) * 128; }

    const v4f* __restrict__ trow4 = (const v4f*)(T + (size_t)g * DIMF);
    const v4f  tv   = trow4[lane];
    const float bias = bscal[0];

    __shared__ float lg[128];

    const int per = L >> 2;                 // logits per wave: 16 (k4) or 32 (k8)
    const int j0  = wave * per;
    #pragma unroll 2
    for (int j = j0; j < j0 + per; ++j) {
        const int edge = ebase[j];
        const v4f ev = ((const v4f*)(emat + (size_t)edge * DIMF))[lane];
        float p = ev.x * tv.x + ev.y * tv.y + ev.z * tv.z + ev.w * tv.w;
        // wave32 butterfly reduction
        p += __shfl_xor(p, 16, 32);
        p += __shfl_xor(p,  8, 32);
        p += __shfl_xor(p,  4, 32);
        p += __shfl_xor(p,  2, 32);
        p += __shfl_xor(p,  1, 32);
        if (lane == 0) {
            const float logit = p + bias;
            const float sg    = 1.0f / (1.0f + expf(-logit));
            lg[j] = expf(sg / TAU_F);
        }
    }
    __syncthreads();

    if (tid == 0) {
        float s[8];
        #pragma unroll
        for (int c = 0; c < 8; ++c) {
            if (c < k) {
                float a = 0.0f;
                for (int d = 0; d < 16; ++d) a += lg[c * 16 + d];
                s[c] = a;
            } else {
                s[c] = 0.0f;
            }
        }
        float suf[8];
        float run = 0.0f;
        for (int c = k - 1; c >= 0; --c) { run += s[c]; suf[c] = run; }
        float lsum = 0.0f;
        for (int c = 0; c < k - 1; ++c) {
            float ratio = s[c] / suf[c];
            ratio = fminf(ratio, BETA_F);
            lsum += -logf(ratio);
        }
        lossv[g] = lsum / (float)(k - 1);
    }
}

// ---------------------------------------------------------------------------
// Kernel 3: deterministic mean of 18000 losses. 1 block, 256 threads.
// ---------------------------------------------------------------------------
__global__ void __launch_bounds__(256)
milnce_reduce(const float* __restrict__ lossv, float* __restrict__ out)
{
    __shared__ float sm[256];
    const int t = threadIdx.x;
    float acc = 0.0f;
    for (int i = t; i < NUM_G; i += 256) acc += lossv[i];
    sm[t] = acc;
    __syncthreads();
    for (int s = 128; s > 0; s >>= 1) {
        if (t < s) sm[t] += sm[t + s];
        __syncthreads();
    }
    if (t == 0) out[0] = sm[0] / (float)NUM_G;
}

// ---------------------------------------------------------------------------
extern "C" void kernel_launch(void* const* d_in, const int* in_sizes, int n_in,
                              void* d_out, int out_size, void* d_ws, size_t ws_size,
                              hipStream_t stream)
{
    (void)in_sizes; (void)n_in; (void)out_size; (void)ws_size;

    const float* nmat = (const float*)d_in[0];   // [200000,128]
    const float* emat = (const float*)d_in[1];   // [200000,128]
    const float* Wmat = (const float*)d_in[2];   // [1,128,128]
    const float* bsc  = (const float*)d_in[3];   // [1]
    const int*   hn4  = (const int*)d_in[4];     // [12000]
    const int*   he4  = (const int*)d_in[5];     // [12000,4,16]
    const int*   hn8  = (const int*)d_in[6];     // [6000]
    const int*   he8  = (const int*)d_in[7];     // [6000,8,16]

    float* T     = (float*)d_ws;                 // 18000*128 floats = 9.216 MB
    float* lossv = T + (size_t)NUM_G * DIMF;     // 18000 floats

    milnce_gemm_t_wmma<<<NUM_G / 16, 256, 0, stream>>>(nmat, Wmat, hn4, hn8, T);
    milnce_loss<<<NUM_G, 128, 0, stream>>>(T, emat, bsc, he4, he8, lossv);
    milnce_reduce<<<1, 256, 0, stream>>>(lossv, (float*)d_out);
}